// graph_cnnNet_18966575579432
// MI455X (gfx1250) — compile-verified
//
#include <hip/hip_runtime.h>
#include <hip/hip_bf16.h>

typedef float v2f __attribute__((ext_vector_type(2)));
typedef float v8f __attribute__((ext_vector_type(8)));

#define BATCH 128
#define V0 4096
#define V1 1024
#define E0 32768
#define E1 8192
#define KCH 25
#define KPAD 28
#define C1OUT 32
#define C2IN 32
#define C2OUT 64
#define C2 (C2IN * BATCH)      // 4096 columns in conv2 cheby buffers
#define FC1IN 16384
#define FC1OUT 512

static __device__ __forceinline__ void atomic_fadd(float* p, float v) {
    __hip_atomic_fetch_add(p, v, __ATOMIC_RELAXED, __HIP_MEMORY_SCOPE_AGENT);
}

// ---------------- utility kernels ----------------

__global__ void k_zero(float* p, int n) {
    int i = blockIdx.x * blockDim.x + threadIdx.x;
    if (i < n) p[i] = 0.f;
}

// broadcast bias: h[..., c] = bias[c], c fastest dim of period C
__global__ void k_init_bias(float* h, const float* bias, int n, int C) {
    int i = blockIdx.x * blockDim.x + threadIdx.x;
    if (i < n) h[i] = bias[i % C];
}

// pad cl1_w (32,25) -> (32,28) with zeros
__global__ void k_pad_w1(const float* __restrict__ w, float* __restrict__ wp) {
    int i = blockIdx.x * blockDim.x + threadIdx.x;
    if (i >= C1OUT * KPAD) return;
    int fo = i / KPAD, k = i % KPAD;
    wp[i] = (k < KCH) ? w[fo * KCH + k] : 0.f;
}

// x (B, V) -> T0 (V, B)
__global__ void k_transpose_x(const float* __restrict__ x, float* __restrict__ T) {
    int i = blockIdx.x * blockDim.x + threadIdx.x;
    if (i >= V0 * BATCH) return;
    int v = i / BATCH, b = i % BATCH;
    T[i] = x[b * V0 + v];
}

// p1 (B, V1, 32) -> T (V1, fin*128 + b)
__global__ void k_transpose_p1(const float* __restrict__ p1, float* __restrict__ T) {
    int i = blockIdx.x * blockDim.x + threadIdx.x;
    if (i >= V1 * C2) return;
    int v = i / C2;
    int rest = i % C2;
    int fin = rest / BATCH;
    int b = rest % BATCH;
    T[i] = p1[b * (V1 * C2IN) + v * C2IN + fin];
}

// edge-parallel SpMM scatter: y[rows[e], c] += vals[e] * z[cols[e], c]
__global__ void k_spmm_edges(const int* __restrict__ rows, const int* __restrict__ cols,
                             const float* __restrict__ vals,
                             const float* __restrict__ z, float* __restrict__ y, int C) {
    int e = blockIdx.x;
    int c = blockIdx.y * blockDim.x + threadIdx.x;
    if (c >= C) return;
    int r = rows[e];
    int cl = cols[e];
    float v = vals[e];
    atomic_fadd(&y[r * C + c], v * z[cl * C + c]);
}

// Chebyshev combine: tout = ca*y + cb*tm1 + cc*tm2   (2/LMAX == 1)
__global__ void k_combine(const float* __restrict__ y, const float* __restrict__ tm1,
                          const float* __restrict__ tm2, float* __restrict__ tout,
                          float ca, float cb, float cc, int n) {
    int i = blockIdx.x * blockDim.x + threadIdx.x;
    if (i < n) tout[i] = ca * y[i] + cb * tm1[i] + cc * tm2[i];
}

// relu(max-pool p=4 over v): h (B, Vout*4, C) -> p (B, Vout, C)
__global__ void k_pool_relu(const float* __restrict__ h, float* __restrict__ p,
                            int Vout, int C) {
    int i = blockIdx.x * blockDim.x + threadIdx.x;
    int n = BATCH * Vout * C;
    if (i >= n) return;
    int c = i % C;
    int rest = i / C;
    int vo = rest % Vout;
    int b = rest / Vout;
    const float* src = h + (size_t)b * Vout * 4 * C + (size_t)vo * 4 * C + c;
    float m = src[0];
    m = fmaxf(m, src[C]);
    m = fmaxf(m, src[2 * C]);
    m = fmaxf(m, src[3 * C]);
    p[i] = fmaxf(m, 0.f);
}

// ---------------- WMMA GEMMs (f32 16x16x4) ----------------
// A layout: lane l: row m = l%16, VGPR j holds K = kk + 2*(l/16) + j
// B layout: lane l: col n = l%16, VGPR j holds K = kk + 2*(l/16) + j
// D layout: VGPR r, lane l: M = r + 8*(l/16), N = l%16

// conv1: h1[b, v, fo] = sum_k Tall[k][v][b] * W1p[fo][k] + bias[fo]
// Tall has KPAD=28 slots (25..27 zeroed); W1p is (32,28) zero-padded.
// one block per v; 16 waves = 8 Mtiles x 2 Ntiles
__global__ void k_conv1_gemm(const float* __restrict__ Tall, const float* __restrict__ W,
                             const float* __restrict__ bias, float* __restrict__ h1) {
    int v = blockIdx.x;
    int wave = threadIdx.y;
    int lane = threadIdx.x;
    int half = lane >> 4, l16 = lane & 15;
    int tileM = wave & 7;
    int tileN = wave >> 3;
    v8f c = {0.f, 0.f, 0.f, 0.f, 0.f, 0.f, 0.f, 0.f};
    for (int kk = 0; kk < KPAD; kk += 4) {
        v2f a, b;
#pragma unroll
        for (int j = 0; j < 2; j++) {
            int k = kk + 2 * half + j;
            a[j] = Tall[(size_t)k * (V0 * BATCH) + v * BATCH + tileM * 16 + l16];
            b[j] = W[(tileN * 16 + l16) * KPAD + k];
        }
        c = __builtin_amdgcn_wmma_f32_16x16x4_f32(false, a, false, b, (short)0, c, false, false);
    }
    int fo = tileN * 16 + l16;
    float bv = bias[fo];
#pragma unroll
    for (int r = 0; r < 8; r++) {
        int m = r + 8 * half;
        h1[(size_t)(tileM * 16 + m) * (V0 * C1OUT) + (size_t)v * C1OUT + fo] = c[r] + bv;
    }
}

// conv2 accumulate term k: h2[b, v, fo] += sum_fin Tk[v][fin*128+b] * W2[fo][fin*25+k]
// one block per v; 16 waves x 2 tiles = 8 Mtiles x 4 Ntiles
__global__ void k_conv2_gemm_acc(const float* __restrict__ Tk, const float* __restrict__ W2,
                                 float* __restrict__ h2, int k) {
    int v = blockIdx.x;
    int wave = threadIdx.y;
    int lane = threadIdx.x;
    int half = lane >> 4, l16 = lane & 15;
#pragma unroll
    for (int t = 0; t < 2; t++) {
        int tile = wave * 2 + t;
        int tileM = tile & 7;
        int tileN = tile >> 3;
        int fo = tileN * 16 + l16;
        v8f c;
#pragma unroll
        for (int r = 0; r < 8; r++) {
            int m = r + 8 * half;
            c[r] = h2[(size_t)(tileM * 16 + m) * (V1 * C2OUT) + (size_t)v * C2OUT + fo];
        }
        for (int kk = 0; kk < C2IN; kk += 4) {
            v2f a, b;
#pragma unroll
            for (int j = 0; j < 2; j++) {
                int fin = kk + 2 * half + j;
                a[j] = Tk[(size_t)v * C2 + fin * BATCH + tileM * 16 + l16];
                b[j] = W2[(tileN * 16 + l16) * 800 + fin * KCH + k];
            }
            c = __builtin_amdgcn_wmma_f32_16x16x4_f32(false, a, false, b, (short)0, c, false, false);
        }
#pragma unroll
        for (int r = 0; r < 8; r++) {
            int m = r + 8 * half;
            h2[(size_t)(tileM * 16 + m) * (V1 * C2OUT) + (size_t)v * C2OUT + fo] = c[r];
        }
    }
}

// fc1: out[b, o] += sum_{k in chunk} A[b, k] * W[o, k]; K split into 8 chunks of 2048
// grid (8 Mtiles, 32 Ntiles, 8 Kchunks), 1 wave per block
__global__ void k_fc1_gemm(const float* __restrict__ A, const float* __restrict__ W,
                           float* __restrict__ out) {
    int tileM = blockIdx.x, tileN = blockIdx.y, kc = blockIdx.z;
    int lane = threadIdx.x;
    int half = lane >> 4, l16 = lane & 15;
    v8f c = {0.f, 0.f, 0.f, 0.f, 0.f, 0.f, 0.f, 0.f};
    int kbase = kc * 2048;
    for (int kk = 0; kk < 2048; kk += 4) {
        v2f a, b;
#pragma unroll
        for (int j = 0; j < 2; j++) {
            int k = kbase + kk + 2 * half + j;
            a[j] = A[(size_t)(tileM * 16 + l16) * FC1IN + k];
            b[j] = W[(size_t)(tileN * 16 + l16) * FC1IN + k];
        }
        c = __builtin_amdgcn_wmma_f32_16x16x4_f32(false, a, false, b, (short)0, c, false, false);
    }
    int o = tileN * 16 + l16;
#pragma unroll
    for (int r = 0; r < 8; r++) {
        int m = r + 8 * half;
        atomic_fadd(&out[(size_t)(tileM * 16 + m) * FC1OUT + o], c[r]);
    }
}

// fc2: out[b, o] = sum_j relu(h[b, j]) * W[o, j] + bias[o]
__global__ void k_fc2(const float* __restrict__ h, const float* __restrict__ W,
                      const float* __restrict__ bias, float* __restrict__ out) {
    int i = blockIdx.x * blockDim.x + threadIdx.x;
    if (i >= BATCH * 10) return;
    int b = i / 10, o = i % 10;
    float s = bias[o];
    const float* hr = h + (size_t)b * FC1OUT;
    const float* wr = W + (size_t)o * FC1OUT;
    for (int j = 0; j < FC1OUT; j++) s += fmaxf(hr[j], 0.f) * wr[j];
    out[i] = s;
}

// ---------------- launcher ----------------

extern "C" void kernel_launch(void* const* d_in, const int* in_sizes, int n_in,
                              void* d_out, int out_size, void* d_ws, size_t ws_size,
                              hipStream_t stream) {
    const float* x = (const float*)d_in[0];
    const int* L0r = (const int*)d_in[2];
    const int* L0c = (const int*)d_in[3];
    const float* L0v = (const float*)d_in[4];
    const int* L2r = (const int*)d_in[5];
    const int* L2c = (const int*)d_in[6];
    const float* L2v = (const float*)d_in[7];
    const float* cl1_w = (const float*)d_in[8];
    const float* cl1_b = (const float*)d_in[9];
    const float* cl2_w = (const float*)d_in[10];
    const float* cl2_b = (const float*)d_in[11];
    const float* fc1_w = (const float*)d_in[12];
    const float* fc1_b = (const float*)d_in[13];
    const float* fc2_w = (const float*)d_in[14];
    const float* fc2_b = (const float*)d_in[15];

    float* ws = (float*)d_ws;
    const int V0B = V0 * BATCH;        // 524,288
    const int NC2 = V1 * C2;           // 4,194,304

    // region A [0 .. 14,680,064): Tall1 (28 slots) during conv1; h2/p2/fc1o after
    float* Tall1 = ws;                                   // 28*524288 floats
    float* h2 = ws;                                      // 8,388,608 floats
    float* p2 = ws + 8388608;                            // 2,097,152 floats
    float* fc1o = ws + 8388608 + 2097152;                // 65,536 floats
    // region B [14,680,064 .. 31,457,280): h1 during conv1; T2 rotation + y2 after
    float* h1 = ws + 14680064;                           // 16,777,216 floats
    float* T2buf[3];
    T2buf[0] = ws + 14680064;                            // 4,194,304 floats each
    T2buf[1] = T2buf[0] + 4194304;
    T2buf[2] = T2buf[1] + 4194304;
    float* y2 = T2buf[2] + 4194304;                      // 4,194,304 floats
    // region C: p1
    float* p1 = ws + 31457280;                           // 4,194,304 floats
    // region D: y1
    float* y1 = ws + 31457280 + 4194304;                 // 524,288 floats
    // region E: padded conv1 weights (32 x 28)
    float* W1p = ws + 31457280 + 4194304 + 524288;       // 896 floats

    // ---- conv1 Chebyshev recurrence: T_k into Tall1[k] ----
    k_transpose_x<<<(V0B + 255) / 256, 256, 0, stream>>>(x, Tall1);
    // zero the pad slots k=25..27 and build padded weights
    k_zero<<<(3 * V0B + 255) / 256, 256, 0, stream>>>(Tall1 + (size_t)KCH * V0B, 3 * V0B);
    k_pad_w1<<<(C1OUT * KPAD + 255) / 256, 256, 0, stream>>>(cl1_w, W1p);
    for (int k = 1; k < KCH; k++) {
        const float* Tm1 = Tall1 + (size_t)(k - 1) * V0B;
        const float* Tm2 = Tall1 + (size_t)((k >= 2) ? (k - 2) : 0) * V0B;
        float* Tout = Tall1 + (size_t)k * V0B;
        k_zero<<<(V0B + 255) / 256, 256, 0, stream>>>(y1, V0B);
        k_spmm_edges<<<dim3(E0, 1), 128, 0, stream>>>(L0r, L0c, L0v, Tm1, y1, BATCH);
        if (k == 1)
            k_combine<<<(V0B + 255) / 256, 256, 0, stream>>>(y1, Tm1, Tm2, Tout, 1.f, -1.f, 0.f, V0B);
        else
            k_combine<<<(V0B + 255) / 256, 256, 0, stream>>>(y1, Tm1, Tm2, Tout, 2.f, -2.f, -1.f, V0B);
    }
    // conv1 GEMM (WMMA): 4096 blocks x 16 waves
    k_conv1_gemm<<<V0, dim3(32, 16), 0, stream>>>(Tall1, W1p, cl1_b, h1);
    // relu + pool -> p1 (B, 1024, 32)
    k_pool_relu<<<(BATCH * V1 * C1OUT + 255) / 256, 256, 0, stream>>>(h1, p1, V1, C1OUT);

    // ---- conv2 ----
    k_transpose_p1<<<(NC2 + 255) / 256, 256, 0, stream>>>(p1, T2buf[0]);
    k_init_bias<<<(BATCH * V1 * C2OUT + 255) / 256, 256, 0, stream>>>(h2, cl2_b,
                                                                      BATCH * V1 * C2OUT, C2OUT);
    k_conv2_gemm_acc<<<V1, dim3(32, 16), 0, stream>>>(T2buf[0], cl2_w, h2, 0);
    for (int k = 1; k < KCH; k++) {
        const float* Tm1 = T2buf[(k - 1) % 3];
        const float* Tm2 = T2buf[(k >= 2) ? ((k - 2) % 3) : ((k - 1) % 3)];
        float* Tout = T2buf[k % 3];
        k_zero<<<(NC2 + 255) / 256, 256, 0, stream>>>(y2, NC2);
        k_spmm_edges<<<dim3(E1, C2 / 256), 256, 0, stream>>>(L2r, L2c, L2v, Tm1, y2, C2);
        if (k == 1)
            k_combine<<<(NC2 + 255) / 256, 256, 0, stream>>>(y2, Tm1, Tm2, Tout, 1.f, -1.f, 0.f, NC2);
        else
            k_combine<<<(NC2 + 255) / 256, 256, 0, stream>>>(y2, Tm1, Tm2, Tout, 2.f, -2.f, -1.f, NC2);
        k_conv2_gemm_acc<<<V1, dim3(32, 16), 0, stream>>>(Tout, cl2_w, h2, k);
    }
    // relu + pool -> p2 (B, 256, 64) == flat (128, 16384)
    k_pool_relu<<<(BATCH * 256 * C2OUT + 255) / 256, 256, 0, stream>>>(h2, p2, 256, C2OUT);

    // ---- FC1 (WMMA, K split 8 ways, atomic f32 accumulate) ----
    k_init_bias<<<(BATCH * FC1OUT + 255) / 256, 256, 0, stream>>>(fc1o, fc1_b, BATCH * FC1OUT,
                                                                  FC1OUT);
    k_fc1_gemm<<<dim3(8, 32, 8), 32, 0, stream>>>(p2, fc1_w, fc1o);

    // ---- FC2 -> d_out ----
    k_fc2<<<(BATCH * 10 + 127) / 128, 128, 0, stream>>>(fc1o, fc2_w, fc2_b, (float*)d_out);
}